// SequenceGRU_64433099374751
// MI455X (gfx1250) — compile-verified
//
#include <hip/hip_runtime.h>
#include <cmath>

#define HH   51
#define G3   153
#define HP   52        // padded K (multiple of 4)
#define GP   160       // padded gate columns (10 x 16)
#define TLEN 1024
#define BATCH 2048
#define KSTEPS 13      // 52 / 4

typedef float v2f __attribute__((ext_vector_type(2)));
typedef float v8f __attribute__((ext_vector_type(8)));

__device__ __forceinline__ float sig_(float x) { return 1.0f / (1.0f + expf(-x)); }

__global__ void __launch_bounds__(64)
gru2_scan_kernel(const float* __restrict__ input,
                 const float* __restrict__ w_ih1,
                 const float* __restrict__ w_hh1,
                 const float* __restrict__ b_ih1,
                 const float* __restrict__ b_hh1,
                 const float* __restrict__ w_ih2,
                 const float* __restrict__ w_hh2,
                 const float* __restrict__ b_ih2,
                 const float* __restrict__ b_hh2,
                 const float* __restrict__ lin_w,
                 const float* __restrict__ lin_b,
                 float* __restrict__ out)
{
    // LDS: padded weights (B-matrix layout-friendly: row n contiguous in k),
    // hidden states, gate buffers, biases.
    __shared__ float sW[3][GP * HP];   // 0: Whh1 (+x-fold col), 1: Wih2, 2: Whh2
    __shared__ float sG[2][16 * GP];   // gate scratch (gi2 / gh2)
    __shared__ float sH[2][16 * HP];   // h1, h2 (col 51 of h1 carries x_t)
    __shared__ float sBi1[GP], sBh1[GP], sBi2[GP], sBh2[GP], sWi1[GP], sLw[HP];

    const int tid   = threadIdx.x;
    const int Rbase = blockIdx.x * 16;

    // ---------------- one-time init ----------------
    for (int i = tid; i < GP * HP; i += 64) {
        const int n = i / HP, k = i - n * HP;
        float whh1 = 0.f, wih2 = 0.f, whh2 = 0.f;
        if (n < G3) {
            if (k < HH) {
                whh1 = w_hh1[n * HH + k];
                wih2 = w_ih2[n * HH + k];
                whh2 = w_hh2[n * HH + k];
            } else {               // k == 51: fold x*w_ih1 into r,z gates only
                whh1 = (n < 2 * HH) ? w_ih1[n] : 0.f;
            }
        }
        sW[0][i] = whh1; sW[1][i] = wih2; sW[2][i] = whh2;
    }
    for (int i = tid; i < GP; i += 64) {
        sBi1[i] = (i < G3) ? b_ih1[i] : 0.f;
        sBh1[i] = (i < G3) ? b_hh1[i] : 0.f;
        sBi2[i] = (i < G3) ? b_ih2[i] : 0.f;
        sBh2[i] = (i < G3) ? b_hh2[i] : 0.f;
        sWi1[i] = (i < G3) ? w_ih1[i] : 0.f;
    }
    for (int i = tid; i < HP; i += 64) sLw[i] = (i < HH) ? lin_w[i] : 0.f;
    for (int i = tid; i < 16 * HP; i += 64) { sH[0][i] = 0.f; sH[1][i] = 0.f; }
    __syncthreads();

    const int wv   = tid >> 5;     // wave id (0/1): N-tile split
    const int lane = tid & 31;
    const int j    = lane & 15;    // column-in-tile / A row
    const int half = lane >> 4;    // lane half selects K/M offsets
    const float linb = lin_b[0];

    for (int t = 0; t < TLEN; ++t) {
        // ---- preA: stage x_t into h1's padded column ----
        if (tid < 16) sH[0][tid * HP + HH] = input[(Rbase + tid) * TLEN + t];
        __syncthreads();

        // ---- A: gh1 = [h1 | x] @ [Whh1 | wih1_rz]^T -> sG[0] ----
        for (int nt = wv * 5; nt < wv * 5 + 5; ++nt) {
            v8f acc = {0.f, 0.f, 0.f, 0.f, 0.f, 0.f, 0.f, 0.f};
            const float* wb = &sW[0][(nt * 16 + j) * HP + 2 * half];
            const float* ha = &sH[0][j * HP + 2 * half];
#pragma unroll
            for (int ks = 0; ks < KSTEPS; ++ks) {
                v2f a = *(const v2f*)(ha + ks * 4);
                v2f b = *(const v2f*)(wb + ks * 4);
                acc = __builtin_amdgcn_wmma_f32_16x16x4_f32(
                          false, a, false, b, (short)0, acc, false, false);
            }
#pragma unroll
            for (int v = 0; v < 8; ++v)
                sG[0][(v + 8 * half) * GP + nt * 16 + j] = acc[v];
        }
        __syncthreads();

        // ---- B: layer-1 elementwise -> h1 ----
        for (int idx = tid; idx < 16 * HH; idx += 64) {
            const int row = idx / HH, c = idx - row * HH;
            const float x  = sH[0][row * HP + HH];
            float r = sG[0][row * GP + c]          + sBi1[c]          + sBh1[c];
            float z = sG[0][row * GP + HH + c]     + sBi1[HH + c]     + sBh1[HH + c];
            r = sig_(r); z = sig_(z);
            const float i_n = x * sWi1[2 * HH + c] + sBi1[2 * HH + c];
            const float h_n = sG[0][row * GP + 2 * HH + c] + sBh1[2 * HH + c];
            const float n   = tanhf(i_n + r * h_n);
            const float ho  = sH[0][row * HP + c];
            sH[0][row * HP + c] = (1.f - z) * n + z * ho;
        }
        __syncthreads();

        // ---- C: gi2 = h1 @ Wih2^T -> sG[0]; gh2 = h2 @ Whh2^T -> sG[1] ----
        for (int nt = wv * 5; nt < wv * 5 + 5; ++nt) {
            v8f acc0 = {0.f, 0.f, 0.f, 0.f, 0.f, 0.f, 0.f, 0.f};
            v8f acc1 = {0.f, 0.f, 0.f, 0.f, 0.f, 0.f, 0.f, 0.f};
            const float* wb1 = &sW[1][(nt * 16 + j) * HP + 2 * half];
            const float* wb2 = &sW[2][(nt * 16 + j) * HP + 2 * half];
            const float* h1a = &sH[0][j * HP + 2 * half];
            const float* h2a = &sH[1][j * HP + 2 * half];
#pragma unroll
            for (int ks = 0; ks < KSTEPS; ++ks) {
                v2f a0 = *(const v2f*)(h1a + ks * 4);
                v2f b0 = *(const v2f*)(wb1 + ks * 4);
                acc0 = __builtin_amdgcn_wmma_f32_16x16x4_f32(
                           false, a0, false, b0, (short)0, acc0, false, false);
                v2f a1 = *(const v2f*)(h2a + ks * 4);
                v2f b1 = *(const v2f*)(wb2 + ks * 4);
                acc1 = __builtin_amdgcn_wmma_f32_16x16x4_f32(
                           false, a1, false, b1, (short)0, acc1, false, false);
            }
#pragma unroll
            for (int v = 0; v < 8; ++v) {
                sG[0][(v + 8 * half) * GP + nt * 16 + j] = acc0[v];
                sG[1][(v + 8 * half) * GP + nt * 16 + j] = acc1[v];
            }
        }
        __syncthreads();

        // ---- D: layer-2 elementwise -> h2 ----
        for (int idx = tid; idx < 16 * HH; idx += 64) {
            const int row = idx / HH, c = idx - row * HH;
            float r = sG[0][row * GP + c]      + sBi2[c]
                    + sG[1][row * GP + c]      + sBh2[c];
            float z = sG[0][row * GP + HH + c] + sBi2[HH + c]
                    + sG[1][row * GP + HH + c] + sBh2[HH + c];
            r = sig_(r); z = sig_(z);
            const float i_n = sG[0][row * GP + 2 * HH + c] + sBi2[2 * HH + c];
            const float h_n = sG[1][row * GP + 2 * HH + c] + sBh2[2 * HH + c];
            const float n   = tanhf(i_n + r * h_n);
            const float ho  = sH[1][row * HP + c];
            sH[1][row * HP + c] = (1.f - z) * n + z * ho;
        }
        __syncthreads();

        // ---- E: output projection out[b,t] = h2 . lin_w + lin_b ----
        if (tid < 16) {
            float s = linb;
            const float* hr = &sH[1][tid * HP];
#pragma unroll
            for (int c = 0; c < HH; ++c) s += hr[c] * sLw[c];
            out[(Rbase + tid) * TLEN + t] = s;
        }
        // next preA is wave0-only writes; barrier after preA orders them for all.
    }
}

extern "C" void kernel_launch(void* const* d_in, const int* in_sizes, int n_in,
                              void* d_out, int out_size, void* d_ws, size_t ws_size,
                              hipStream_t stream) {
    (void)in_sizes; (void)n_in; (void)d_ws; (void)ws_size; (void)out_size;
    const float* input = (const float*)d_in[0];
    const float* w_ih1 = (const float*)d_in[1];
    const float* w_hh1 = (const float*)d_in[2];
    const float* b_ih1 = (const float*)d_in[3];
    const float* b_hh1 = (const float*)d_in[4];
    const float* w_ih2 = (const float*)d_in[5];
    const float* w_hh2 = (const float*)d_in[6];
    const float* b_ih2 = (const float*)d_in[7];
    const float* b_hh2 = (const float*)d_in[8];
    const float* lin_w = (const float*)d_in[9];
    const float* lin_b = (const float*)d_in[10];
    // d_in[11] is `future` == 0; ignored.

    dim3 grid(BATCH / 16);
    dim3 block(64);
    gru2_scan_kernel<<<grid, block, 0, stream>>>(
        input, w_ih1, w_hh1, b_ih1, b_hh1,
        w_ih2, w_hh2, b_ih2, b_hh2, lin_w, lin_b, (float*)d_out);
}